// Model_69879117906313
// MI455X (gfx1250) — compile-verified
//
#include <hip/hip_runtime.h>

// Mamba-2 SSD chunked scan for gfx1250 (MI455X).
// fp32 WMMA 16x16x4, b64 LDS operand fetches, float4 staging, TDM state load.
// b=4, L=4096, h=32, p=64, n=16, chunk=64, chunks=64.

typedef float v2f __attribute__((ext_vector_type(2)));
typedef float v8f __attribute__((ext_vector_type(8)));
typedef unsigned int u32x4 __attribute__((ext_vector_type(4)));
typedef int i32x4 __attribute__((ext_vector_type(4)));
typedef int i32x8 __attribute__((ext_vector_type(8)));

static constexpr int Bsz = 4, Lsz = 4096, Hsz = 32, Psz = 64, Nsz = 16;
static constexpr int CL = 64;                 // chunk length
static constexpr int NC = Lsz / CL;           // 64 chunks
static constexpr int TILES = Bsz * NC * Hsz;  // 8192
static constexpr int XS = 66;                 // padded stride for [*][l/s] K-major tiles
static constexpr int NS = 18;                 // padded stride for [*][n] / [*][s16] tiles
static constexpr long STATE_ELEMS = (long)Bsz * Hsz * NC * Psz * Nsz; // 8,388,608 floats

#ifndef __has_builtin
#define __has_builtin(x) 0
#endif
#if defined(__HIP_DEVICE_COMPILE__) && __has_builtin(__builtin_amdgcn_tensor_load_to_lds)
#define USE_TDM 1
#else
#define USE_TDM 0
#endif

__device__ __forceinline__ v8f wmma_f32(v2f a, v2f b, v8f c) {
    // D = A(16x4 f32) * B(4x16 f32) + C(16x16 f32)
    return __builtin_amdgcn_wmma_f32_16x16x4_f32(
        false, a, false, b, (short)0, c, false, false);
}
__device__ __forceinline__ v2f lds2(const float* p) {
    // K-pair fetch: one aligned ds_load_b64
    return *(const v2f*)__builtin_assume_aligned(p, 8);
}

// ---------------------------------------------------------------------------
// Kernel 1: per-chunk states  S_c[p][n] = sum_l X[l][p] * exp(cum[63]-cum[l]) * B[l][n]
// One WG per (b,c,h); 128 threads = 4 waves; wave w owns p-tile [16w,16w+16).
// LDS layouts are K(=l)-major so every WMMA operand pair is a b64 load.
// ---------------------------------------------------------------------------
__global__ __launch_bounds__(128) void k1_states(
    const float* __restrict__ X, const float* __restrict__ A,
    const float* __restrict__ Bm, float* __restrict__ states,
    float* __restrict__ asum)
{
    __shared__ __align__(16) float sXT[Psz * XS];   // X^T: [p][l]
    __shared__ __align__(16) float sBT[Nsz * XS];   // (B*decay)^T: [n][l]
    __shared__ __align__(16) float sCum[CL];
    __shared__ __align__(16) float sA[CL];

    const int tid = threadIdx.x;
    const int idx = blockIdx.x;
    const int hh = idx & 31;
    const int cc = (idx >> 5) & 63;
    const int bb = idx >> 11;
    const long lbase = (long)bb * Lsz + (long)cc * CL;

    const float4* X4 = reinterpret_cast<const float4*>(X);
    for (int i = tid; i < CL * Psz / 4; i += 128) {     // 1024 float4s
        int l = i >> 4, p4 = (i & 15) * 4;
        float4 v = X4[((lbase + l) * Hsz + hh) * (Psz / 4) + (p4 >> 2)];
        sXT[(p4 + 0) * XS + l] = v.x;
        sXT[(p4 + 1) * XS + l] = v.y;
        sXT[(p4 + 2) * XS + l] = v.z;
        sXT[(p4 + 3) * XS + l] = v.w;
    }
    if (tid < CL) sA[tid] = A[(lbase + tid) * Hsz + hh];
    __syncthreads();
    if (tid == 0) {
        float r = 0.f;
        for (int l = 0; l < CL; ++l) { r += sA[l]; sCum[l] = r; }
        asum[(bb * Hsz + hh) * NC + cc] = r;   // chunk total for inter-chunk scan
    }
    __syncthreads();
    const float ctot = sCum[CL - 1];
    const float4* B4 = reinterpret_cast<const float4*>(Bm);
    for (int i = tid; i < CL * Nsz / 4; i += 128) {     // 256 float4s
        int l = i >> 2, n4 = (i & 3) * 4;
        float d = __expf(ctot - sCum[l]);
        float4 v = B4[((lbase + l) * Hsz + hh) * (Nsz / 4) + (n4 >> 2)];
        sBT[(n4 + 0) * XS + l] = v.x * d;
        sBT[(n4 + 1) * XS + l] = v.y * d;
        sBT[(n4 + 2) * XS + l] = v.z * d;
        sBT[(n4 + 3) * XS + l] = v.w * d;
    }
    __syncthreads();

    const int w = tid >> 5, lane = tid & 31;
    const int half = lane >> 4, mn = lane & 15;
    const int prow = 16 * w + mn;

    v8f acc = {};
    for (int kb = 0; kb < CL; kb += 4) {    // K = l, 16 WMMA steps
        const int k0 = kb + 2 * half;       // even -> 8B aligned
        acc = wmma_f32(lds2(&sXT[prow * XS + k0]),   // X^T[p][l]
                       lds2(&sBT[mn * XS + k0]),     // Bd[l][n] (K-major)
                       acc);
    }

    const long sbase = ((long)(bb * Hsz + hh) * NC + cc) * (Psz * Nsz);
    for (int v = 0; v < 8; ++v) {
        int p = 16 * w + v + 8 * half;      // D layout: M = v + 8*(lane>=16)
        states[sbase + p * Nsz + mn] = acc[v];
    }
}

// ---------------------------------------------------------------------------
// Kernel 2: inter-chunk sequential scan (in place, float4 vectorized).
//   prior_0 = 0;  write prior_z over states_z;  S <- exp(a_z)*S + chunk_state_z
// ---------------------------------------------------------------------------
__global__ __launch_bounds__(256) void k2_scan(
    float* __restrict__ states, const float* __restrict__ asum)
{
    const int bh = blockIdx.x;              // 0..127
    float4* st4 = reinterpret_cast<float4*>(states);
    const long base4 = (long)bh * NC * (Psz * Nsz / 4);
    const float* as = asum + bh * NC;

    float4 S; S.x = 0.f; S.y = 0.f; S.z = 0.f; S.w = 0.f;
    for (int z = 0; z < NC; ++z) {
        const float dz = __expf(as[z]);
        const long i = base4 + (long)z * (Psz * Nsz / 4) + threadIdx.x;
        float4 c = st4[i];
        st4[i] = S;
        S.x = dz * S.x + c.x; S.y = dz * S.y + c.y;
        S.z = dz * S.z + c.z; S.w = dz * S.w + c.w;
    }
}

// ---------------------------------------------------------------------------
// Kernel 3: Y = Y_diag + Y_off per (b,c,h) tile.
// 512 threads = 16 waves; wave (rt,ct) owns output tile l in [16rt,..), p in [16ct,..).
// Prior state comes in via the Tensor Data Mover, padded by TDM into the
// stride-18 LDS layout the WMMA b64 reads want.
// ---------------------------------------------------------------------------
__global__ __launch_bounds__(512) void k3_output(
    const float* __restrict__ X, const float* __restrict__ A,
    const float* __restrict__ Bm, const float* __restrict__ Cm,
    const float* __restrict__ states, float* __restrict__ out)
{
    __shared__ __align__(16) float sXT[Psz * XS];      // X^T: [p][s]
    __shared__ __align__(16) float sB[CL * NS];        // [s][n], padded
    __shared__ __align__(16) float sC[CL * NS];        // [l][n], padded
    __shared__ __align__(16) float sS[Psz * NS];       // prior state [p][n], padded
    __shared__ __align__(16) float sG[16 * 16 * NS];   // per-wave (G∘L) scratch
    __shared__ __align__(16) float sCum[CL];
    __shared__ __align__(16) float sA[CL];

    const int tid = threadIdx.x;
    const int idx = blockIdx.x;
    const int hh = idx & 31;
    const int cc = (idx >> 5) & 63;
    const int bb = idx >> 11;
    const long lbase = (long)bb * Lsz + (long)cc * CL;
    const long sbase = ((long)(bb * Hsz + hh) * NC + cc) * (Psz * Nsz);

#if USE_TDM
    if (tid == 0) {
        // D# group0: count=1 | lds_addr | 57-bit global addr | type=2
        const unsigned lds_addr =
            (unsigned)(unsigned long long)(const void*)&sS[0]; // addr[31:0] = LDS offset
        const unsigned long long ga =
            (unsigned long long)(const void*)(states + sbase);
        u32x4 g0;
        g0[0] = 1u;
        g0[1] = lds_addr;
        g0[2] = (unsigned)ga;
        g0[3] = ((unsigned)(ga >> 32) & 0x1FFFFFFu) | (2u << 30);
        // D# group1: data_size=4B; pad every 16 dwords by 2 dwords (-> stride 18);
        // 1-D tile of 1024 elements.
        i32x8 g1;
        g1[0] = (int)((2u << 16) | (1u << 20) | (3u << 22) | (1u << 25));
        g1[1] = (int)(1024u << 16);   // tensor_dim0[15:0]
        g1[2] = (int)(1u << 16);      // tensor_dim1 = 1
        g1[3] = (int)(1024u << 16);   // tile_dim0 = 1024
        g1[4] = 0;                    // tile_dim1/2 unused
        g1[5] = 1024;                 // tensor_dim0_stride
        g1[6] = 0;
        g1[7] = 0;
        i32x4 z4 = {};
        i32x8 z8 = {};
        // 6-arg form on this toolchain: (g0, g1, g2, g3, g4, cpol)
        __builtin_amdgcn_tensor_load_to_lds(g0, g1, z4, z4, z8, 0);
    }
#else
    for (int i = tid; i < Psz * Nsz; i += 512)
        sS[(i >> 4) * NS + (i & 15)] = states[sbase + i];
#endif

    const float4* X4 = reinterpret_cast<const float4*>(X);
    for (int i = tid; i < CL * Psz / 4; i += 512) {     // 1024 float4s
        int l = i >> 4, p4 = (i & 15) * 4;
        float4 v = X4[((lbase + l) * Hsz + hh) * (Psz / 4) + (p4 >> 2)];
        sXT[(p4 + 0) * XS + l] = v.x;
        sXT[(p4 + 1) * XS + l] = v.y;
        sXT[(p4 + 2) * XS + l] = v.z;
        sXT[(p4 + 3) * XS + l] = v.w;
    }
    const float4* B4 = reinterpret_cast<const float4*>(Bm);
    const float4* C4 = reinterpret_cast<const float4*>(Cm);
    for (int i = tid; i < CL * Nsz / 4; i += 512) {     // 256 float4s
        int l = i >> 2, n4 = (i & 3) * 4;
        long g = ((lbase + l) * Hsz + hh) * (Nsz / 4) + (n4 >> 2);
        float4 vb = B4[g], vc = C4[g];
        sB[l * NS + n4 + 0] = vb.x; sB[l * NS + n4 + 1] = vb.y;
        sB[l * NS + n4 + 2] = vb.z; sB[l * NS + n4 + 3] = vb.w;
        sC[l * NS + n4 + 0] = vc.x; sC[l * NS + n4 + 1] = vc.y;
        sC[l * NS + n4 + 2] = vc.z; sC[l * NS + n4 + 3] = vc.w;
    }
    if (tid < CL) sA[tid] = A[(lbase + tid) * Hsz + hh];
#if USE_TDM
    __builtin_amdgcn_s_wait_tensorcnt(0);   // issuing wave drains TDM before barrier
#endif
    __syncthreads();
    if (tid == 0) {
        float r = 0.f;
        for (int l = 0; l < CL; ++l) { r += sA[l]; sCum[l] = r; }
    }
    __syncthreads();

    const int w = tid >> 5, lane = tid & 31;
    const int half = lane >> 4, mn = lane & 15;
    const int rt = w >> 2, ct = w & 3;
    const int lrow = 16 * rt + mn;          // A-operand row (M = l)
    const int pcol = 16 * ct + mn;          // N column (p)
    const float dl = __expf(sCum[lrow]);    // state_decay_out for this A row

    v8f acc = {};

    // Y_off = (C * exp(cum)) @ S^T      (K = n = 16 -> 4 WMMA)
    for (int kb = 0; kb < Nsz; kb += 4) {
        const int k0 = kb + 2 * half;
        v2f a = lds2(&sC[lrow * NS + k0]) * dl;
        v2f b = lds2(&sS[pcol * NS + k0]);     // S^T[n][p]
        acc = wmma_f32(a, b, acc);
    }

    float* gw = sG + w * (16 * NS);         // wave-private scratch
    for (int st = 0; st < 4; ++st) {
        // G tile = C @ B^T over n        (K = 16 -> 4 WMMA)
        v8f g = {};
        const int scol = 16 * st + mn;
        for (int kb = 0; kb < Nsz; kb += 4) {
            const int k0 = kb + 2 * half;
            g = wmma_f32(lds2(&sC[lrow * NS + k0]),
                         lds2(&sB[scol * NS + k0]),   // B^T[n][s]
                         g);
        }
        // Apply causal decay mask L[l][s] = exp(cum[l]-cum[s]) * (l>=s);
        // park in LDS row-major (stride 18) for A-layout re-read.
        for (int v = 0; v < 8; ++v) {
            const int lr = v + 8 * half;              // D layout local row
            const int lg = 16 * rt + lr;
            const int sg = 16 * st + mn;
            const float val = (lg >= sg) ? g[v] * __expf(sCum[lg] - sCum[sg]) : 0.f;
            gw[lr * NS + mn] = val;
        }
        // acc += (G∘L) @ X over s        (K = 16 -> 4 WMMA)
        for (int kb = 0; kb < 16; kb += 4) {
            const int k0 = kb + 2 * half;
            v2f a = lds2(&gw[mn * NS + k0]);                       // (G∘L)[l][s]
            v2f b = lds2(&sXT[pcol * XS + 16 * st + k0]);          // X[s][p]
            acc = wmma_f32(a, b, acc);
        }
    }

    for (int v = 0; v < 8; ++v) {
        const int lg = 16 * rt + v + 8 * half;
        out[((lbase + lg) * Hsz + hh) * Psz + pcol] = acc[v];
    }
}

// ---------------------------------------------------------------------------
extern "C" void kernel_launch(void* const* d_in, const int* in_sizes, int n_in,
                              void* d_out, int out_size, void* d_ws, size_t ws_size,
                              hipStream_t stream) {
    const float* X = (const float*)d_in[0];
    const float* A = (const float*)d_in[1];
    const float* B = (const float*)d_in[2];
    const float* C = (const float*)d_in[3];
    float* out = (float*)d_out;

    float* states = (float*)d_ws;                 // 32 MiB: [b][h][c][p][n]
    float* asum   = states + STATE_ELEMS;         // 8192 floats: [b][h][c]

    k1_states<<<TILES, 128, 0, stream>>>(X, A, B, states, asum);
    k2_scan<<<Bsz * Hsz, 256, 0, stream>>>(states, asum);
    k3_output<<<TILES, 512, 0, stream>>>(X, A, B, C, states, out);
}